// KANLayer_76819785056464
// MI455X (gfx1250) — compile-verified
//
#include <hip/hip_runtime.h>

// KAN layer: IN_F=256, OUT_F=256, NUM_BASIS=16, BATCH=1024, beta=16
// Outputs (flat concat): y[1024*256], preact[1024*256*256],
//                        postact[...], postspline[...]

typedef float v2f __attribute__((ext_vector_type(2)));
typedef float v4f __attribute__((ext_vector_type(4)));
typedef float v8f __attribute__((ext_vector_type(8)));

#define IN_F   256
#define OUT_F  256
#define NB     16
#define BATCH  1024
#define BETA   16.0f

__global__ __launch_bounds__(256) void kan_wmma_kernel(
    const float* __restrict__ x,        // [1024,256]
    const float* __restrict__ bw,       // [16,256]  basis_weights
    const float* __restrict__ bb,       // [16,256]  basis_bias
    const float* __restrict__ W,        // [256, 16*256]
    const float* __restrict__ bias,     // [256]
    const float* __restrict__ sb,       // [256,256] scale_base
    float* __restrict__ y,              // [1024,256]
    float* __restrict__ preact,         // [1024,256,256]
    float* __restrict__ postact,        // [1024,256,256]
    float* __restrict__ postspline)     // [1024,256,256]
{
    const int nt     = blockIdx.x;        // out tile   (16 tiles of 16)
    const int bt     = blockIdx.y;        // batch tile (64 tiles of 16)
    const int lane   = threadIdx.x & 31;
    const int wave   = threadIdx.x >> 5;  // 8 waves
    const int laneLo = lane & 15;
    const int laneHi = lane >> 4;

    const int kg = nt * 16 + laneLo;      // this lane's output column (B/C: N = lane%16)
    const float kb = bias[kg] * (1.0f / OUT_F);

    v8f yacc = {0.f, 0.f, 0.f, 0.f, 0.f, 0.f, 0.f, 0.f};

    __shared__ float red[8 * 256];

    // Each wave processes 8-consecutive-i chunks; 32 chunks cover i = 0..255.
    for (int c = wave; c < 32; c += 8) {
        const int i0 = c * 8;
        v8f ps[8];   // postspline tile values, one v8f (C-fragment) per i
        v8f pa[8];   // postact tile values

        // scale_base for this lane's column, 8 consecutive i (coalesced b128 loads)
        const v4f sb_lo = *(const v4f*)(sb + (size_t)kg * IN_F + i0);
        const v4f sb_hi = *(const v4f*)(sb + (size_t)kg * IN_F + i0 + 4);
        const float sbv[8] = {sb_lo.x, sb_lo.y, sb_lo.z, sb_lo.w,
                              sb_hi.x, sb_hi.y, sb_hi.z, sb_hi.w};

        #pragma unroll
        for (int j = 0; j < 8; ++j) {
            const int i = i0 + j;

            // x for A-fragment rows: A-matrix M = lane%16
            const float xv = x[((size_t)(bt * 16 + laneLo)) * IN_F + i];

            v8f acc = {0.f, 0.f, 0.f, 0.f, 0.f, 0.f, 0.f, 0.f};
            // K=16 contraction over basis dim as 4 chained 16x16x4 f32 WMMAs.
            // Fragment f covers K (=b) in [4f, 4f+4); per-lane slots:
            //   b0 = 4f + 2*(lane>>4), b1 = b0+1  (identical enumeration for A and B)
            #pragma unroll
            for (int f = 0; f < 4; ++f) {
                const int b0 = 4 * f + 2 * laneHi;
                const int b1 = b0 + 1;

                const float t0 = fmaf(xv, bw[b0 * IN_F + i], bb[b0 * IN_F + i]);
                const float t1 = fmaf(xv, bw[b1 * IN_F + i], bb[b1 * IN_F + i]);

                v2f afrag;
                afrag.x = __expf(-BETA * t0 * t0);   // gaussian RBF basis
                afrag.y = __expf(-BETA * t1 * t1);

                v2f bfrag;
                bfrag.x = W[(size_t)kg * (NB * IN_F) + b0 * IN_F + i];
                bfrag.y = W[(size_t)kg * (NB * IN_F) + b1 * IN_F + i];

                acc = __builtin_amdgcn_wmma_f32_16x16x4_f32(
                          false, afrag, false, bfrag, (short)0, acc, false, false);
            }
            ps[j] = acc;

            // postact = postspline + cos(x[M,i])*scale_base[k,i] + bias[k]/OUT_F
            // C layout: row M = r + 8*(lane>>4); broadcast cos(x) from lane M.
            const int cxbits = __float_as_int(__cosf(xv));
            v8f p;
            #pragma unroll
            for (int r = 0; r < 8; ++r) {
                const int mr = r + 8 * laneHi;
                const float cxm =
                    __int_as_float(__builtin_amdgcn_ds_bpermute(4 * mr, cxbits));
                p[r] = acc[r] + cxm * sbv[j] + kb;
            }
            pa[j] = p;
            yacc += p;
        }

        // Stores: for each C-row r, 8 consecutive-i values -> two b128 stores
        // per tensor (32B granules per lane along the contiguous i axis).
        #pragma unroll
        for (int r = 0; r < 8; ++r) {
            const int Mg = bt * 16 + r + 8 * laneHi;
            const size_t base = (((size_t)Mg * OUT_F) + kg) * IN_F + i0;

            v4f lo, hi;
            lo = (v4f){ps[0][r], ps[1][r], ps[2][r], ps[3][r]};
            hi = (v4f){ps[4][r], ps[5][r], ps[6][r], ps[7][r]};
            *(v4f*)(postspline + base)     = lo;
            *(v4f*)(postspline + base + 4) = hi;

            lo = (v4f){pa[0][r], pa[1][r], pa[2][r], pa[3][r]};
            hi = (v4f){pa[4][r], pa[5][r], pa[6][r], pa[7][r]};
            *(v4f*)(postact + base)     = lo;
            *(v4f*)(postact + base + 4) = hi;

            // preact = broadcast of x: reload row directly (L0/L2 hit)
            lo = *(const v4f*)(x + (size_t)Mg * IN_F + i0);
            hi = *(const v4f*)(x + (size_t)Mg * IN_F + i0 + 4);
            *(v4f*)(preact + base)     = lo;
            *(v4f*)(preact + base + 4) = hi;
        }
    }

    // Deterministic cross-wave reduction of y tile through LDS.
    #pragma unroll
    for (int r = 0; r < 8; ++r)
        red[wave * 256 + lane * 8 + r] = yacc[r];
    __syncthreads();

    {
        const int t = threadIdx.x;       // element index 0..255 (== lane*8 + r)
        const int l = t >> 3;            // original lane
        const int r = t & 7;             // original C-fragment VGPR row
        float s = 0.0f;
        #pragma unroll
        for (int w2 = 0; w2 < 8; ++w2)
            s += red[w2 * 256 + t];      // fixed order -> deterministic
        const int Mg  = bt * 16 + r + 8 * (l >> 4);
        const int kg2 = nt * 16 + (l & 15);
        y[(size_t)Mg * OUT_F + kg2] = s;
    }
}

extern "C" void kernel_launch(void* const* d_in, const int* in_sizes, int n_in,
                              void* d_out, int out_size, void* d_ws, size_t ws_size,
                              hipStream_t stream) {
    (void)in_sizes; (void)n_in; (void)d_ws; (void)ws_size; (void)out_size;

    const float* x    = (const float*)d_in[0];
    const float* bw   = (const float*)d_in[1];
    const float* bb   = (const float*)d_in[2];
    const float* W    = (const float*)d_in[3];
    const float* bias = (const float*)d_in[4];
    const float* sb   = (const float*)d_in[5];

    float* y          = (float*)d_out;
    float* preact     = y + (size_t)BATCH * OUT_F;                       // 262144
    float* postact    = preact + (size_t)BATCH * OUT_F * IN_F;           // +67108864
    float* postspline = postact + (size_t)BATCH * OUT_F * IN_F;          // +67108864

    dim3 grid(OUT_F / 16, BATCH / 16);   // (16, 64) = 1024 blocks
    dim3 block(256);                      // 8 wave32
    kan_wmma_kernel<<<grid, block, 0, stream>>>(x, bw, bb, W, bias, sb,
                                                y, preact, postact, postspline);
}